// LMSolver_45518063403435
// MI455X (gfx1250) — compile-verified
//
#include <hip/hip_runtime.h>

typedef __attribute__((ext_vector_type(2))) float v2f;
typedef __attribute__((ext_vector_type(8))) float v8f;

#define NUM_ITER 10
#define LM_EPS   1e-5f
#define NPTS     512
#define TPB      32             // single wave32 per pose -> no barriers
#define PPT      (NPTS / TPB)   // 16 points per thread
#define NACC     14             // upper triangle of 5x5 gram [J | r], minus (4,4), incl dead (0,1)

// Accumulator index map (i,j) of gram G = [J|r]^T [J|r], col 4 = residual:
//  0:(0,0) 1:(0,1)=0 2:(0,2) 3:(0,3) 4:(0,4)
//  5:(1,1) 6:(1,2) 7:(1,3) 8:(1,4)
//  9:(2,2) 10:(2,3) 11:(2,4)
// 12:(3,3) 13:(3,4)

__device__ __forceinline__ float fast_rcp(float x) {
    float r = __builtin_amdgcn_rcpf(x);       // v_rcp_f32
    float e = fmaf(-x, r, 1.0f);              // 1 Newton-Raphson step -> ~1 ulp
    return fmaf(r, e, r);
}

// Partial row reduction with fused DPP adds: after row_shr 1,2,4 lanes
// 7/15/23/31 hold the sums of lanes 0-7 / 8-15 / 16-23 / 24-31
// (bound_ctrl:1 -> out-of-row source lanes read 0). The remaining 4-way
// reduction is done by the WMMA (K=4).
__device__ __forceinline__ float dpp_row_sum8(float s) {
    float t;
    asm("v_add_f32_dpp %0, %1, %1 row_shr:1 row_mask:0xf bank_mask:0xf bound_ctrl:1"
        : "=v"(t) : "v"(s));
    asm("v_add_f32_dpp %0, %1, %1 row_shr:2 row_mask:0xf bank_mask:0xf bound_ctrl:1"
        : "=v"(s) : "v"(t));
    asm("v_add_f32_dpp %0, %1, %1 row_shr:4 row_mask:0xf bank_mask:0xf bound_ctrl:1"
        : "=v"(t) : "v"(s));
    return t;
}

__device__ __forceinline__ float readlane_f32(float v, int srclane) {
    return __int_as_float(__builtin_amdgcn_readlane(__float_as_int(v), srclane));
}

__global__ __launch_bounds__(TPB) void lm4dof_kernel(
    const float* __restrict__ x3d,
    const float* __restrict__ x2d,
    const float* __restrict__ w2d,
    const float* __restrict__ cam,
    const float* __restrict__ pose_init,
    float* __restrict__ pose_out)
{
    const int b = blockIdx.x;
    const int t = threadIdx.x;      // == lane (single wave32)

    __shared__ float sPart[4 * 16]; // 4 eighth-wave partial slots = WMMA K-slots

    // One-time zero (covers the never-written pad entries 14/15 of each slot).
    sPart[t] = 0.0f;
    sPart[32 + t] = 0.0f;

    const float* x3b = x3d + (size_t)b * NPTS * 3;
    const float* x2b = x2d + (size_t)b * NPTS * 2;
    const float* w2b = w2d + (size_t)b * NPTS * 2;

    const float fx = cam[4 * b + 0], fy = cam[4 * b + 1];
    const float cx = cam[4 * b + 2], cy = cam[4 * b + 3];

    // --- one-time load of this pose's point data into registers, with
    //     camera/weight constants pre-folded per point ---
    float X0[PPT], X1[PPT], X2[PPT];
    float WUFX[PPT], WVFY[PPT];   // wu*fx, wv*fy
    float WUCU[PPT], WVCV[PPT];   // wu*(cx - u_obs), wv*(cy - v_obs)
#pragma unroll
    for (int p = 0; p < PPT; ++p) {
        const int i = t + p * TPB;
        X0[p] = x3b[3 * i + 0];
        X1[p] = x3b[3 * i + 1];
        X2[p] = x3b[3 * i + 2];
        const float uo = x2b[2 * i + 0];
        const float vo = x2b[2 * i + 1];
        const float wu = w2b[2 * i + 0];
        const float wvw = w2b[2 * i + 1];
        WUFX[p] = wu * fx;
        WVFY[p] = wvw * fy;
        WUCU[p] = wu * (cx - uo);
        WVCV[p] = wvw * (cy - vo);
    }

    // Pose lives in registers, uniform across all lanes.
    float tx  = pose_init[4 * b + 0];
    float ty  = pose_init[4 * b + 1];
    float tz  = pose_init[4 * b + 2];
    float yaw = pose_init[4 * b + 3];

    for (int it = 0; it < NUM_ITER; ++it) {
        const float cth = __cosf(yaw);
        const float sth = __sinf(yaw);

        float acc[NACC];
#pragma unroll
        for (int k = 0; k < NACC; ++k) acc[k] = 0.0f;

        // --- per-point residual + jacobian + gram accumulation (pure VALU fp32) ---
#pragma unroll
        for (int p = 0; p < PPT; ++p) {
            const float px   = fmaf(cth, X0[p], fmaf(sth, X2[p], tx));
            const float py   = X1[p] + ty;
            const float pz   = fmaf(-sth, X0[p], fmaf(cth, X2[p], tz));
            const float invz = fast_rcp(pz);
            const float pxz  = px * invz;
            const float pyz  = py * invz;
            // ru = wu*(fx*px/pz + cx - u_obs), folded: fma(wu*fx, pxz, wu*(cx-u))
            const float ru = fmaf(WUFX[p], pxz, WUCU[p]);
            const float rv = fmaf(WVFY[p], pyz, WVCV[p]);
            const float dpx = pz - tz;        // = -s*X0 + c*X2
            const float dpz = tx - px;        // = -c*X0 - s*X2
            const float ju0 = WUFX[p] * invz;
            const float ju2 = -ju0 * pxz;
            const float ju3 = ju0 * fmaf(-pxz, dpz, dpx);
            const float jv1 = WVFY[p] * invz;
            const float jv2 = -jv1 * pyz;
            const float jv3 = jv2 * dpz;

            acc[0]  = fmaf(ju0, ju0, acc[0]);
            acc[2]  = fmaf(ju0, ju2, acc[2]);
            acc[3]  = fmaf(ju0, ju3, acc[3]);
            acc[4]  = fmaf(ju0, ru,  acc[4]);
            acc[5]  = fmaf(jv1, jv1, acc[5]);
            acc[6]  = fmaf(jv1, jv2, acc[6]);
            acc[7]  = fmaf(jv1, jv3, acc[7]);
            acc[8]  = fmaf(jv1, rv,  acc[8]);
            acc[9]  = fmaf(ju2, ju2, fmaf(jv2, jv2, acc[9]));
            acc[10] = fmaf(ju2, ju3, fmaf(jv2, jv3, acc[10]));
            acc[11] = fmaf(ju2, ru,  fmaf(jv2, rv,  acc[11]));
            acc[12] = fmaf(ju3, ju3, fmaf(jv3, jv3, acc[12]));
            acc[13] = fmaf(ju3, ru,  fmaf(jv3, rv,  acc[13]));
        }

        // --- eighth-wave reduction on the VALU pipe (3 fused DPP adds each);
        //     lanes 7/15/23/31 hold the four partials ---
#pragma unroll
        for (int k = 0; k < NACC; ++k) {
            if (k == 1) continue;       // (0,1) entry is identically zero
            acc[k] = dpp_row_sum8(acc[k]);
        }
        if ((t & 7) == 7) {
            const int slot = t >> 3;    // 0..3
#pragma unroll
            for (int k = 0; k < NACC; ++k) sPart[slot * 16 + k] = acc[k];
        }
        // Same wave: LDS ops are in-order, no barrier needed.

        // --- final 4-way reduction via V_WMMA_F32_16X16X4_F32 (EXEC = all 1s) ---
        // D[m,n] = sum_{k=0..3} A[m,k] * 1.0 ; A[m,k] = partial slot k, entry m.
        // A fragment (16x4 f32): lanes 0-15 hold K=0/.x, K=1/.y; lanes 16-31
        // hold K=2/.x, K=3/.y; M = lane % 16.
        {
            const int m  = t & 15;
            const int kk = (t >> 4) << 1;   // 0 or 2
            v2f a;
            a.x = sPart[(kk + 0) * 16 + m];
            a.y = sPart[(kk + 1) * 16 + m];
            v2f ones;
            ones.x = 1.0f;
            ones.y = 1.0f;
            v8f d = {};
            d = __builtin_amdgcn_wmma_f32_16x16x4_f32(false, a, false, ones,
                                                      (short)0, d, false, false);

            // C/D layout: vgpr v holds row v at lanes 0-15 and row v+8 at
            // lanes 16-31; all columns are identical (B = ones). Broadcast the
            // 14 gram entries to every lane with v_readlane (no LDS traffic).
            float s[NACC];
#pragma unroll
            for (int k = 0; k < 8; ++k) s[k] = readlane_f32(d[k], 0);
#pragma unroll
            for (int k = 0; k < 6; ++k) s[8 + k] = readlane_f32(d[k], 16);

            // --- 4x4 SPD solve (Gauss-Jordan, damped diagonal), computed
            //     redundantly by all lanes (uniform) -> pose stays in registers ---
            float A[4][4];
            float g[4];
            A[0][0] = s[0] + LM_EPS;
            A[0][1] = s[1];           A[1][0] = s[1];
            A[0][2] = s[2];           A[2][0] = s[2];
            A[0][3] = s[3];           A[3][0] = s[3];
            A[1][1] = s[5] + LM_EPS;
            A[1][2] = s[6];           A[2][1] = s[6];
            A[1][3] = s[7];           A[3][1] = s[7];
            A[2][2] = s[9] + LM_EPS;
            A[2][3] = s[10];          A[3][2] = s[10];
            A[3][3] = s[12] + LM_EPS;
            g[0] = s[4];
            g[1] = s[8];
            g[2] = s[11];
            g[3] = s[13];

#pragma unroll
            for (int k = 0; k < 4; ++k) {
                const float inv = fast_rcp(A[k][k]);
#pragma unroll
                for (int j = 0; j < 4; ++j) A[k][j] *= inv;
                g[k] *= inv;
#pragma unroll
                for (int i = 0; i < 4; ++i) {
                    if (i != k) {
                        const float f = A[i][k];
#pragma unroll
                        for (int j = 0; j < 4; ++j) A[i][j] -= f * A[k][j];
                        g[i] -= f * g[k];
                    }
                }
            }
            tx  -= g[0];
            ty  -= g[1];
            tz  -= g[2];
            yaw -= g[3];
        }
    }

    if (t == 0) {
        float4 out;
        out.x = tx;
        out.y = ty;
        out.z = tz;
        out.w = yaw;
        *(float4*)(pose_out + 4 * b) = out;
    }
}

extern "C" void kernel_launch(void* const* d_in, const int* in_sizes, int n_in,
                              void* d_out, int out_size, void* d_ws, size_t ws_size,
                              hipStream_t stream) {
    const float* x3d       = (const float*)d_in[0];
    const float* x2d       = (const float*)d_in[1];
    const float* w2d       = (const float*)d_in[2];
    const float* cam       = (const float*)d_in[3];
    const float* pose_init = (const float*)d_in[4];
    float* out             = (float*)d_out;

    const int B = in_sizes[4] / 4;   // pose_init is (B,4)
    lm4dof_kernel<<<B, TPB, 0, stream>>>(x3d, x2d, w2d, cam, pose_init, out);
}